// SimplifiedTwoStageGNN_18597208392464
// MI455X (gfx1250) — compile-verified
//
#include <hip/hip_runtime.h>
#include <hip/hip_bf16.h>

// ---------------------------------------------------------------------------
// Two-stage GNN for MI455X (gfx1250, wave32).
// GEMMs use V_WMMA_F32_16X16X4_F32 (full fp32 precision, matches reference).
// Edge aggregation uses global fp32 atomics (node features fit in 192MB L2).
// ---------------------------------------------------------------------------

#define NN 262144
#define EE 4194304
#define PP 8192
#define CC 256
#define EEI 32768
#define FF 89
#define HH 64

typedef __attribute__((ext_vector_type(2))) float v2f;
typedef __attribute__((ext_vector_type(8))) float v8f;

// ---------------- small elementwise kernels ----------------

__global__ __launch_bounds__(256) void fill_f32(float* p, float v, int n) {
  int i = blockIdx.x * blockDim.x + threadIdx.x;
  if (i < n) p[i] = v;
}

__global__ __launch_bounds__(256) void deg_accum(const int* __restrict__ dst,
                                                 float* __restrict__ deg, int n) {
  int e = blockIdx.x * blockDim.x + threadIdx.x;
  if (e < n) atomicAdd(&deg[dst[e]], 1.0f);
}

__global__ __launch_bounds__(256) void rsqrt_inplace(float* p, int n) {
  int i = blockIdx.x * blockDim.x + threadIdx.x;
  if (i < n) p[i] = __frsqrt_rn(p[i]);
}

// h[i][c] *= dinv[i]  (produces hs in place)
__global__ __launch_bounds__(256) void scale_rows(float* __restrict__ h,
                                                  const float* __restrict__ dinv,
                                                  int nodes) {
  int i = blockIdx.x * blockDim.x + threadIdx.x;
  if (i < nodes * HH) h[i] *= dinv[i >> 6];
}

// one wave (32 lanes) per edge; each lane moves 2 channels (64 total)
__global__ __launch_bounds__(256) void edge_scatter(const float* __restrict__ hs,
                                                    const int* __restrict__ src,
                                                    const int* __restrict__ dst,
                                                    float* __restrict__ agg, int nedges) {
  long long t = (long long)blockIdx.x * blockDim.x + threadIdx.x;
  int e = (int)(t >> 5);
  int lane = (int)(t & 31);
  if (e >= nedges) return;
  int s = src[e];
  int d = dst[e];
  const float2 v = *(const float2*)(hs + (size_t)s * HH + lane * 2);
  float* pd = agg + (size_t)d * HH + lane * 2;
  atomicAdd(pd + 0, v.x);
  atomicAdd(pd + 1, v.y);
}

// out[i][c] = relu(dinv[i]*(agg[i][c] + hs[i][c]) + b[c])
__global__ __launch_bounds__(256) void gcn_finalize(const float* __restrict__ agg,
                                                    const float* __restrict__ hs,
                                                    const float* __restrict__ dinv,
                                                    const float* __restrict__ bias,
                                                    float* __restrict__ out, int nodes) {
  int i = blockIdx.x * blockDim.x + threadIdx.x;
  if (i >= nodes * HH) return;
  int node = i >> 6, c = i & 63;
  float v = dinv[node] * (agg[i] + hs[i]) + bias[c];
  out[i] = v > 0.0f ? v : 0.0f;
}

// mean over 32 consecutive rows (batch = arange(n)//32 in the reference)
__global__ __launch_bounds__(256) void mean_pool32(const float* __restrict__ h,
                                                   float* __restrict__ out, int ngroups) {
  int i = blockIdx.x * blockDim.x + threadIdx.x;
  if (i >= ngroups * HH) return;
  int g = i >> 6, c = i & 63;
  const float* p = h + (size_t)g * 32 * HH + c;
  float s = 0.0f;
#pragma unroll
  for (int r = 0; r < 32; ++r) s += p[r * HH];
  out[i] = s * (1.0f / 32.0f);
}

__global__ __launch_bounds__(256) void bias_relu64(float* __restrict__ t,
                                                   const float* __restrict__ b, int rows) {
  int i = blockIdx.x * blockDim.x + threadIdx.x;
  if (i >= rows * HH) return;
  float v = t[i] + b[i & 63];
  t[i] = v > 0.0f ? v : 0.0f;
}

__global__ __launch_bounds__(256) void head_kernel(const float* __restrict__ t,
                                                   const float* __restrict__ Wp2,
                                                   const float* __restrict__ bp2,
                                                   float* __restrict__ out) {
  int i = blockIdx.x * blockDim.x + threadIdx.x;
  if (i >= CC) return;
  float s = bp2[0];
#pragma unroll
  for (int k = 0; k < HH; ++k) s += t[i * HH + k] * Wp2[k];
  out[i] = s;
}

// ---------------- fp32 WMMA GEMM: C[M x 64] = A[M x K] @ W[K x 64] ----------
// Block = 256 threads = 8 waves. Block computes a 32-row x 64-col strip.
// Wave w: row tile = w>>2 (16 rows), col tile = (w&3)*16.
// K padded to Kpad (multiple of 4, <= 96) with zeros in LDS.
//
// Fragment layouts per CDNA5 ISA 7.12.2:
//   A 16x4 fp32 : lanes 0-15 -> M=lane, v0=K0, v1=K1 ; lanes 16-31 -> v0=K2, v1=K3
//   B 4x16 fp32 : lanes 0-15 -> N=lane, v0=K0, v1=K1 ; lanes 16-31 -> v0=K2, v1=K3
//   C/D 16x16   : VGPR r, lanes 0-15 -> M=r ; lanes 16-31 -> M=r+8 ; N=lane&15
__global__ __launch_bounds__(256) void gemm_h64_wmma(const float* __restrict__ A,
                                                     const float* __restrict__ W,
                                                     float* __restrict__ C,
                                                     int M, int K, int Kpad) {
  __shared__ float sW[96][65];  // padded stride: stores/loads conflict-light
  __shared__ float sA[32][97];  // stride 97 -> 16 row-strided lanes hit 16 banks

  const int tid = threadIdx.x;
  const int row_base = blockIdx.x * 32;

  // stage W (zero-padded to 96 rows)
  for (int idx = tid; idx < 96 * 64; idx += 256) {
    int r = idx >> 6, c = idx & 63;
    sW[r][c] = (r < K) ? W[r * 64 + c] : 0.0f;
  }
  // stage A tile (32 x Kpad, zero-padded to 96 cols)
  for (int idx = tid; idx < 32 * 96; idx += 256) {
    int r = idx / 96, c = idx % 96;
    sA[r][c] = (c < K) ? A[(size_t)(row_base + r) * K + c] : 0.0f;
  }
  __syncthreads();

  const int wave = tid >> 5;
  const int lane = tid & 31;
  const int half = lane >> 4;
  const int l16 = lane & 15;
  const int rt = wave >> 2;           // 0..1
  const int col0 = (wave & 3) << 4;   // 0,16,32,48

  v8f acc = {};
  for (int k0 = 0; k0 < Kpad; k0 += 4) {
    const int ka = k0 + 2 * half;
    v2f a, b;
    a.x = sA[rt * 16 + l16][ka];
    a.y = sA[rt * 16 + l16][ka + 1];
    b.x = sW[ka][col0 + l16];
    b.y = sW[ka + 1][col0 + l16];
    acc = __builtin_amdgcn_wmma_f32_16x16x4_f32(false, a, false, b, (short)0, acc,
                                                false, false);
  }

  const int orow = row_base + rt * 16 + half * 8;
#pragma unroll
  for (int r = 0; r < 8; ++r) {
    C[(size_t)(orow + r) * 64 + col0 + l16] = acc[r];
  }
}

// ---------------------------------------------------------------------------

static inline unsigned cdiv(long long a, long long b) { return (unsigned)((a + b - 1) / b); }

extern "C" void kernel_launch(void* const* d_in, const int* in_sizes, int n_in,
                              void* d_out, int out_size, void* d_ws, size_t ws_size,
                              hipStream_t stream) {
  (void)in_sizes; (void)n_in; (void)out_size; (void)ws_size;

  const float* x   = (const float*)d_in[0];
  const int* src   = (const int*)d_in[1];
  const int* dst   = (const int*)d_in[2];
  const int* isrc  = (const int*)d_in[4];
  const int* idst  = (const int*)d_in[5];
  const float* W1  = (const float*)d_in[7];  const float* b1  = (const float*)d_in[8];
  const float* W2  = (const float*)d_in[9];  const float* b2  = (const float*)d_in[10];
  const float* W3  = (const float*)d_in[11]; const float* b3  = (const float*)d_in[12];
  const float* Wi1 = (const float*)d_in[13]; const float* bi1 = (const float*)d_in[14];
  const float* Wi2 = (const float*)d_in[15]; const float* bi2 = (const float*)d_in[16];
  const float* Wp1 = (const float*)d_in[17]; const float* bp1 = (const float*)d_in[18];
  const float* Wp2 = (const float*)d_in[19]; const float* bp2 = (const float*)d_in[20];
  float* out = (float*)d_out;

  // workspace layout
  char* base = (char*)d_ws;
  const size_t nodeMat = (size_t)NN * HH * sizeof(float);   // 64 MB
  const size_t polyMat = (size_t)PP * HH * sizeof(float);   // 2 MB
  float* H  = (float*)(base);                         // gemm output / hs
  float* G  = (float*)(base + nodeMat);               // scatter accumulator
  float* S  = (float*)(base + 2 * nodeMat);           // layer output / next input
  float* dA = (float*)(base + 3 * nodeMat);           // atom dinv [N]
  char* p2  = base + 3 * nodeMat + (size_t)NN * sizeof(float);
  float* PS = (float*)(p2);                            // poly state
  float* PH = (float*)(p2 + polyMat);
  float* PG = (float*)(p2 + 2 * polyMat);
  float* dP = (float*)(p2 + 3 * polyMat);              // poly dinv [P]
  char* p3  = p2 + 3 * polyMat + (size_t)PP * sizeof(float);
  float* CR = (float*)(p3);                            // crystal [256,64]
  float* T  = (float*)(p3 + (size_t)CC * HH * sizeof(float));

  const dim3 blk(256);

  // --- degree -> dinv (self loop included: init 1, add 1 per incoming edge) ---
  fill_f32<<<cdiv(NN, 256), blk, 0, stream>>>(dA, 1.0f, NN);
  deg_accum<<<cdiv(EE, 256), blk, 0, stream>>>(dst, dA, EE);
  rsqrt_inplace<<<cdiv(NN, 256), blk, 0, stream>>>(dA, NN);
  fill_f32<<<cdiv(PP, 256), blk, 0, stream>>>(dP, 1.0f, PP);
  deg_accum<<<cdiv(EEI, 256), blk, 0, stream>>>(idst, dP, EEI);
  rsqrt_inplace<<<cdiv(PP, 256), blk, 0, stream>>>(dP, PP);

  const unsigned gemmN  = NN / 32;           // 8192 blocks
  const unsigned gemmP  = PP / 32;           // 256 blocks
  const unsigned elemN  = cdiv((long long)NN * HH, 256);
  const unsigned elemP  = cdiv((long long)PP * HH, 256);
  const unsigned scatN  = cdiv((long long)EE * 32, 256);
  const unsigned scatP  = cdiv((long long)EEI * 32, 256);

  // --- atom-level GCN layer 1 (K=89 -> pad 96) ---
  gemm_h64_wmma<<<gemmN, blk, 0, stream>>>(x, W1, H, NN, FF, 96);
  scale_rows<<<elemN, blk, 0, stream>>>(H, dA, NN);
  hipMemsetAsync(G, 0, nodeMat, stream);
  edge_scatter<<<scatN, blk, 0, stream>>>(H, src, dst, G, EE);
  gcn_finalize<<<elemN, blk, 0, stream>>>(G, H, dA, b1, S, NN);

  // --- atom-level GCN layers 2,3 (K=64) ---
  gemm_h64_wmma<<<gemmN, blk, 0, stream>>>(S, W2, H, NN, HH, 64);
  scale_rows<<<elemN, blk, 0, stream>>>(H, dA, NN);
  hipMemsetAsync(G, 0, nodeMat, stream);
  edge_scatter<<<scatN, blk, 0, stream>>>(H, src, dst, G, EE);
  gcn_finalize<<<elemN, blk, 0, stream>>>(G, H, dA, b2, S, NN);

  gemm_h64_wmma<<<gemmN, blk, 0, stream>>>(S, W3, H, NN, HH, 64);
  scale_rows<<<elemN, blk, 0, stream>>>(H, dA, NN);
  hipMemsetAsync(G, 0, nodeMat, stream);
  edge_scatter<<<scatN, blk, 0, stream>>>(H, src, dst, G, EE);
  gcn_finalize<<<elemN, blk, 0, stream>>>(G, H, dA, b3, S, NN);

  // --- atoms -> polyhedra mean pool ---
  mean_pool32<<<elemP, blk, 0, stream>>>(S, PS, PP);

  // --- poly-level GCN layers 1,2 ---
  gemm_h64_wmma<<<gemmP, blk, 0, stream>>>(PS, Wi1, PH, PP, HH, 64);
  scale_rows<<<elemP, blk, 0, stream>>>(PH, dP, PP);
  hipMemsetAsync(PG, 0, polyMat, stream);
  edge_scatter<<<scatP, blk, 0, stream>>>(PH, isrc, idst, PG, EEI);
  gcn_finalize<<<elemP, blk, 0, stream>>>(PG, PH, dP, bi1, PS, PP);

  gemm_h64_wmma<<<gemmP, blk, 0, stream>>>(PS, Wi2, PH, PP, HH, 64);
  scale_rows<<<elemP, blk, 0, stream>>>(PH, dP, PP);
  hipMemsetAsync(PG, 0, polyMat, stream);
  edge_scatter<<<scatP, blk, 0, stream>>>(PH, isrc, idst, PG, EEI);
  gcn_finalize<<<elemP, blk, 0, stream>>>(PG, PH, dP, bi2, PS, PP);

  // --- polyhedra -> crystal mean pool ---
  mean_pool32<<<cdiv((long long)CC * HH, 256), blk, 0, stream>>>(PS, CR, CC);

  // --- predictor head ---
  gemm_h64_wmma<<<CC / 32, blk, 0, stream>>>(CR, Wp1, T, CC, HH, 64);
  bias_relu64<<<cdiv((long long)CC * HH, 256), blk, 0, stream>>>(T, bp1, CC);
  head_kernel<<<1, blk, 0, stream>>>(T, Wp2, bp2, out);
}